// BehavioralGNN_87531433492752
// MI455X (gfx1250) — compile-verified
//
#include <hip/hip_runtime.h>
#include <hip/hip_bf16.h>

typedef __attribute__((ext_vector_type(8))) float v8f;
typedef __attribute__((ext_vector_type(2))) float v2f;

// ---------------------------------------------------------------- utilities
__global__ void zero_kernel(float* __restrict__ p, int n) {
  int i = blockIdx.x * blockDim.x + threadIdx.x;
  if (i < n) p[i] = 0.0f;
}

__global__ void degree_kernel(const int* __restrict__ dst, float* __restrict__ deg, int nEdges) {
  int i = blockIdx.x * blockDim.x + threadIdx.x;
  if (i < nEdges)
    __hip_atomic_fetch_add(&deg[dst[i]], 1.0f, __ATOMIC_RELAXED, __HIP_MEMORY_SCOPE_AGENT);
}

__global__ void recip_deg_kernel(float* __restrict__ deg, int n) {
  int i = blockIdx.x * blockDim.x + threadIdx.x;
  if (i < n) deg[i] = 1.0f / fmaxf(deg[i], 1.0f);
}

// One thread per (edge, 4-float chunk): gather x[src] chunk, scatter-add into agg[dst].
// Consecutive lanes cover consecutive chunks of one row -> coalesced float4 gathers.
// Working set (agg + activations + index arrays) is L2-resident (192 MB), so the
// global_atomic_add_f32 stream resolves at L2 rate; this is the roofline term (~2 GB total).
__global__ void scatter_add_kernel(const float* __restrict__ x, const int* __restrict__ src,
                                   const int* __restrict__ dst, float* __restrict__ agg,
                                   int nEdges, int f4) {
  int tid = blockIdx.x * blockDim.x + threadIdx.x;
  int total = nEdges * f4;
  if (tid >= total) return;
  int e = tid / f4;
  int c = tid - e * f4;
  int s = src[e];
  int d = dst[e];
  float4 v = ((const float4*)x)[s * f4 + c];
  float* p = agg + ((size_t)d * f4 + c) * 4;
  __hip_atomic_fetch_add(p + 0, v.x, __ATOMIC_RELAXED, __HIP_MEMORY_SCOPE_AGENT);
  __hip_atomic_fetch_add(p + 1, v.y, __ATOMIC_RELAXED, __HIP_MEMORY_SCOPE_AGENT);
  __hip_atomic_fetch_add(p + 2, v.z, __ATOMIC_RELAXED, __HIP_MEMORY_SCOPE_AGENT);
  __hip_atomic_fetch_add(p + 3, v.w, __ATOMIC_RELAXED, __HIP_MEMORY_SCOPE_AGENT);
}

// ---------------------------------------------------------------- WMMA GEMM
// out[n, o] = act( (agg[n,:]*rdeg[n] + xin[n,:]) @ W[o,:] + bias[o] )
// W is [Fout x FIN] row-major. Block: 128 threads = 4 wave32, 64 rows/block,
// each wave owns a 16-row strip. A-tile AND B-tile staged in LDS (B zero-padded
// past Fout) so the K-loop is pure ds_load_b64 + v_wmma with full EXEC.
template <int FIN>
__global__ __launch_bounds__(128) void
gemm_bias_act_kernel(const float* __restrict__ agg, const float* __restrict__ rdeg,
                     const float* __restrict__ xin, const float* __restrict__ W,
                     const float* __restrict__ bias, float* __restrict__ out,
                     int nRows, int Fout, int doRelu)
{
  constexpr int STRIDE = FIN + 4;          // 16B-aligned rows, conflict-free banks (4*r mod 64)
  __shared__ float As[64 * STRIDE];
  __shared__ float Bs[128 * STRIDE];       // up to 8 col-tiles of 16
  __shared__ float bsh[128];

  const int rowBase   = blockIdx.x * 64;
  const int foutTiles = (Fout + 15) >> 4;
  const int bRows     = foutTiles * 16;

  // Stage B = W^T tile (zero-padded rows past Fout) — all masking happens here, once.
  for (int idx = threadIdx.x; idx < bRows * FIN; idx += 128) {
    int r = idx / FIN;
    int k = idx - r * FIN;
    Bs[r * STRIDE + k] = (r < Fout) ? W[(size_t)r * FIN + k] : 0.0f;
  }
  {
    int t = threadIdx.x;
    if (t < 128) bsh[t] = (t < Fout) ? bias[t] : 0.0f;
  }
  // Stage A = agg/deg + x (heads pass agg==nullptr -> A = x), zero-padded past nRows.
  for (int idx = threadIdx.x; idx < 64 * FIN; idx += 128) {
    int r  = idx / FIN;
    int k  = idx - r * FIN;
    int gr = rowBase + r;
    float v = 0.0f;
    if (gr < nRows) {
      v = xin[(size_t)gr * FIN + k];
      if (agg) v += agg[(size_t)gr * FIN + k] * rdeg[gr];
    }
    As[r * STRIDE + k] = v;
  }
  __syncthreads();

  const int lane = threadIdx.x & 31;
  const int wave = threadIdx.x >> 5;
  const int half = lane >> 4;              // 0 = lanes 0-15, 1 = lanes 16-31
  const int l16  = lane & 15;
  const float* Arow = &As[(wave * 16 + l16) * STRIDE];   // A frag: lane holds row M = l16
  const bool fullRows = (rowBase + 64 <= nRows);         // wave-uniform

  for (int ct = 0; ct < foutTiles; ++ct) {
    const float* Brow = &Bs[(ct * 16 + l16) * STRIDE];   // B frag: lane holds column N = l16
    v8f c = {};
    // V_WMMA_F32_16X16X4_F32: per step, lane supplies K = 2*half + {0,1}
    #pragma unroll 8
    for (int k0 = 0; k0 < FIN; k0 += 4) {
      int ka = k0 + 2 * half;
      v2f a, b;
      a.x = Arow[ka];
      a.y = Arow[ka + 1];
      b.x = Brow[ka];
      b.y = Brow[ka + 1];
      c = __builtin_amdgcn_wmma_f32_16x16x4_f32(false, a, false, b, (short)0, c, false, false);
    }
    const int n  = ct * 16 + l16;
    float bv = bsh[n];
    // D layout: VGPR r -> M = r + 8*half, N = l16
    float* colBase = out + (size_t)(rowBase + wave * 16 + 8 * half) * Fout + n;
    if (fullRows && (ct * 16 + 16 <= Fout)) {
      // Fast path (wave-uniform): 8 unconditional stores, no exec masking.
      #pragma unroll
      for (int r = 0; r < 8; ++r) {
        float val = c[r] + bv;
        if (doRelu) val = fmaxf(val, 0.0f);
        colBase[(size_t)r * Fout] = val;
      }
    } else {
      // Boundary path: last row-block (50000 % 64 = 16) or ragged tile (classes head).
      #pragma unroll
      for (int r = 0; r < 8; ++r) {
        int gRow = rowBase + wave * 16 + r + 8 * half;
        float val = c[r] + bv;
        if (doRelu) val = fmaxf(val, 0.0f);
        if (gRow < nRows && n < Fout) out[(size_t)gRow * Fout + n] = val;
      }
    }
  }
}

// ---------------------------------------------------------------- launch
extern "C" void kernel_launch(void* const* d_in, const int* in_sizes, int n_in,
                              void* d_out, int out_size, void* d_ws, size_t ws_size,
                              hipStream_t stream) {
  const float* x0 = (const float*)d_in[0];
  const int*   ei = (const int*)d_in[1];
  const float *W1 = (const float*)d_in[2],  *b1 = (const float*)d_in[3];
  const float *W2 = (const float*)d_in[4],  *b2 = (const float*)d_in[5];
  const float *W3 = (const float*)d_in[6],  *b3 = (const float*)d_in[7];
  const float *Wp = (const float*)d_in[8],  *bp = (const float*)d_in[9];
  const float *Wc = (const float*)d_in[10], *bc = (const float*)d_in[11];

  const int N      = in_sizes[0] / 64;   // 50000
  const int nEdges = in_sizes[1] / 2;    // 800000
  const int* src = ei;
  const int* dst = ei + nEdges;

  // Workspace layout (floats): deg[N] | agg[N*128] | h2[N*128]  (~51.5 MB)
  float* ws  = (float*)d_ws;
  float* deg = ws;
  float* agg = ws + 65536;
  float* h2  = agg + (size_t)N * 128;

  // d_out layout: emb[N*64] | next[N*64] | classes[N*10].
  // h1 (N*128) is staged inside d_out and dead before layer-3 writes emb.
  float* out = (float*)d_out;
  float* h1  = out;
  float* emb = out;
  float* nxt = out + (size_t)N * 64;
  float* cls = out + (size_t)N * 128;

  auto nb = [](int n) { return dim3((unsigned)((n + 255) / 256)); };
  dim3 b256(256);
  dim3 gemmGrid((unsigned)((N + 63) / 64)), gemmBlock(128);

  // Degrees (shared by all layers): deg -> 1/max(deg,1)
  zero_kernel     <<<nb(N),      b256, 0, stream>>>(deg, N);
  degree_kernel   <<<nb(nEdges), b256, 0, stream>>>(dst, deg, nEdges);
  recip_deg_kernel<<<nb(N),      b256, 0, stream>>>(deg, N);

  // Layer 1: 64 -> 128
  zero_kernel        <<<nb(N * 64),      b256, 0, stream>>>(agg, N * 64);
  scatter_add_kernel <<<nb(nEdges * 16), b256, 0, stream>>>(x0, src, dst, agg, nEdges, 16);
  gemm_bias_act_kernel<64><<<gemmGrid, gemmBlock, 0, stream>>>(agg, deg, x0, W1, b1, h1, N, 128, 1);

  // Layer 2: 128 -> 128
  zero_kernel        <<<nb(N * 128),     b256, 0, stream>>>(agg, N * 128);
  scatter_add_kernel <<<nb(nEdges * 32), b256, 0, stream>>>(h1, src, dst, agg, nEdges, 32);
  gemm_bias_act_kernel<128><<<gemmGrid, gemmBlock, 0, stream>>>(agg, deg, h1, W2, b2, h2, N, 128, 1);

  // Layer 3: 128 -> 64, write node embeddings straight into d_out (h1 dead now)
  zero_kernel        <<<nb(N * 128),     b256, 0, stream>>>(agg, N * 128);
  scatter_add_kernel <<<nb(nEdges * 32), b256, 0, stream>>>(h2, src, dst, agg, nEdges, 32);
  gemm_bias_act_kernel<128><<<gemmGrid, gemmBlock, 0, stream>>>(agg, deg, h2, W3, b3, emb, N, 64, 1);

  // Heads (no aggregation, no relu)
  gemm_bias_act_kernel<64><<<gemmGrid, gemmBlock, 0, stream>>>(nullptr, nullptr, emb, Wp, bp, nxt, N, 64, 0);
  gemm_bias_act_kernel<64><<<gemmGrid, gemmBlock, 0, stream>>>(nullptr, nullptr, emb, Wc, bc, cls, N, 10, 0);
}